// GMM_46188078301710
// MI455X (gfx1250) — compile-verified
//
#include <hip/hip_runtime.h>
#include <math.h>

// ---------------------------------------------------------------------------
// GMM energy on gfx1250 (MI455X), f32 end-to-end via V_WMMA_F32_16X16X4_F32.
//
//  K1: one pass over z,g -> per-workgroup partial Gram/moment tiles
//      ZZ00,ZZ01,ZZ11 (z Gram), Gx0,Gx1 (g x z, with ones-row => Sz),
//      GG (g Gram, ones-row => Sg).                         [WMMA]
//  K2: per batch: reduce partials, mu, cov, Cholesky, A=L^-1, v=A mu, const.
//  K3: per (b,pixel-tile): D = A_k * Z via WMMA (triangular-skip),
//      maha = ||D - v||^2, streaming logsumexp partials.     [WMMA]
//  K4: combine lse partials -> scalar energy.
// ---------------------------------------------------------------------------

typedef float v2f __attribute__((ext_vector_type(2)));
typedef float v8f __attribute__((ext_vector_type(8)));

#define NB   8
#define NC   32
#define NK   5
#define HW   65536
#define EPSF 1e-6f
#define LOG2PI 1.8378770664093453f

__device__ __forceinline__ v8f wmma4(v2f a, v2f b, v8f c) {
  // D(16x16,f32) = A(16x4,f32) * B(4x16,f32) + C
  // 8 args: (neg_a, A, neg_b, B, c_mod, C, reuse_a, reuse_b)
  return __builtin_amdgcn_wmma_f32_16x16x4_f32(false, a, false, b, (short)0, c,
                                               false, false);
}

// ---------------------------------------------------------------------------
// K1: stats pass.  grid = 8 batches x 8 wg, block = 256 (8 waves).
// Each wave: 1024 pixels, macro-iters of 16 px (32B contiguous per lane).
// K-groups of 4 px are {pb+2i, pb+2i+1, pb+8+2i, pb+8+2i+1} so lane halves
// feed the WMMA A/B frag layout directly (reduction order is free).
// ---------------------------------------------------------------------------
__global__ __launch_bounds__(256) void k1_stats(const float* __restrict__ z,
                                                const float* __restrict__ g,
                                                float* __restrict__ part) {
  __shared__ float lds[8 * 6 * 256];
  const int b = blockIdx.x >> 3, wgsub = blockIdx.x & 7;
  const int wave = threadIdx.x >> 5, lane = threadIdx.x & 31;
  const int row = lane & 15, lh = lane >> 4;
  const float* zb = z + (size_t)b * NC * HW;
  const float* gb = g + (size_t)b * NK * HW;
  const int pixbase0 = wgsub * 8192 + wave * 1024;

  v8f dzz00 = {}, dzz01 = {}, dzz11 = {}, dg0 = {}, dg1 = {}, dgg = {};

  for (int it = 0; it < 64; ++it) {
    const int pb = pixbase0 + it * 16;
    const int po = pb + lh * 8;                 // this lane's 8 pixels
    const float* p0 = zb + (size_t)row * HW + po;         // channels 0..15
    const float* p1 = zb + (size_t)(row + 16) * HW + po;  // channels 16..31
    float4 a0 = *(const float4*)p0;
    float4 a1 = *(const float4*)(p0 + 4);
    float4 b0 = *(const float4*)p1;
    float4 b1 = *(const float4*)(p1 + 4);
    float4 g0, g1;
    if (row < NK) {
      const float* pg = gb + (size_t)row * HW + po;
      g0 = *(const float4*)pg;
      g1 = *(const float4*)(pg + 4);
    } else if (row == NK) {             // ones row -> Sz / Sg for free
      g0 = make_float4(1.f, 1.f, 1.f, 1.f); g1 = g0;
    } else {
      g0 = make_float4(0.f, 0.f, 0.f, 0.f); g1 = g0;
    }
    const float za[8]  = {a0.x, a0.y, a0.z, a0.w, a1.x, a1.y, a1.z, a1.w};
    const float zc[8]  = {b0.x, b0.y, b0.z, b0.w, b1.x, b1.y, b1.z, b1.w};
    const float gv[8]  = {g0.x, g0.y, g0.z, g0.w, g1.x, g1.y, g1.z, g1.w};
#pragma unroll
    for (int i = 0; i < 4; ++i) {
      v2f f0 = {za[2 * i], za[2 * i + 1]};
      v2f f1 = {zc[2 * i], zc[2 * i + 1]};
      v2f fg = {gv[2 * i], gv[2 * i + 1]};
      dzz00 = wmma4(f0, f0, dzz00);   // Gram: A-frag == B-frag
      dzz01 = wmma4(f0, f1, dzz01);
      dzz11 = wmma4(f1, f1, dzz11);
      dg0   = wmma4(fg, f0, dg0);     // rows 0-4: gz, row 5: Sz(ch 0-15)
      dg1   = wmma4(fg, f1, dg1);     // rows 0-4: gz, row 5: Sz(ch 16-31)
      dgg   = wmma4(fg, fg, dgg);     // row 5 cols 0-4: Sg
    }
  }

  // fragments -> plain 16x16 matrices in LDS, then tree-reduce 8 waves
  float* my = lds + wave * 1536;
#pragma unroll
  for (int i = 0; i < 8; ++i) {
    const int m = lh * 8 + i, e = m * 16 + row;
    my[0 * 256 + e] = dzz00[i];
    my[1 * 256 + e] = dzz01[i];
    my[2 * 256 + e] = dzz11[i];
    my[3 * 256 + e] = dg0[i];
    my[4 * 256 + e] = dg1[i];
    my[5 * 256 + e] = dgg[i];
  }
  __syncthreads();
  float* out = part + (size_t)blockIdx.x * 1536;
  for (int e = threadIdx.x; e < 1536; e += 256) {
    float s = 0.f;
#pragma unroll
    for (int w = 0; w < 8; ++w) s += lds[w * 1536 + e];
    out[e] = s;
  }
}

// ---------------------------------------------------------------------------
// K2: per-batch finalize. grid = 8, block = 256.  wave k owns component k.
// ---------------------------------------------------------------------------
__global__ __launch_bounds__(256) void k2_prep(const float* __restrict__ part,
                                               float* __restrict__ Aout,
                                               float* __restrict__ vout,
                                               float* __restrict__ cst) {
  __shared__ float red[6 * 256];
  __shared__ float ZZ[1024];
  __shared__ float gz[NK * 32];
  __shared__ float Sz[32];
  __shared__ float Sg[NK];
  __shared__ float mu[NK * 32];
  __shared__ float cv[NK * 1024];
  __shared__ float Ai[NK * 1024];
  const int b = blockIdx.x, tid = threadIdx.x;
  const int kk = tid >> 5, lane = tid & 31;

  for (int e = tid; e < 1536; e += 256) {
    float s = 0.f;
    for (int w = 0; w < 8; ++w) s += part[(size_t)(b * 8 + w) * 1536 + e];
    red[e] = s;
  }
  __syncthreads();
  for (int e = tid; e < 1024; e += 256) {            // assemble ZZ (ZZ10=ZZ01^T)
    const int c = e >> 5, d = e & 31;
    float v;
    if (c < 16 && d < 16)      v = red[0 * 256 + c * 16 + d];
    else if (c < 16)           v = red[1 * 256 + c * 16 + (d - 16)];
    else if (d < 16)           v = red[1 * 256 + d * 16 + (c - 16)];
    else                       v = red[2 * 256 + (c - 16) * 16 + (d - 16)];
    ZZ[e] = v;
  }
  if (tid < NK * 32) {
    const int k = tid >> 5, c = tid & 31;
    gz[tid] = red[(3 + (c >> 4)) * 256 + k * 16 + (c & 15)];
  }
  if (tid < 32) Sz[tid] = red[(3 + (tid >> 4)) * 256 + NK * 16 + (tid & 15)];
  if (tid < NK) Sg[tid] = red[5 * 256 + NK * 16 + tid];
  __syncthreads();

  if (kk < NK) {                                     // mu, l2-normalized
    float m = gz[kk * 32 + lane] / (Sg[kk] + EPSF);
    float n2 = m * m;
    for (int o = 16; o > 0; o >>= 1) n2 += __shfl_xor(n2, o, 32);
    mu[kk * 32 + lane] = m / (EPSF + sqrtf(n2));
  }
  __syncthreads();

  if (kk < NK) {                                     // cov + diag floor + fro
    float fr = 0.f;
    float* cvk = cv + kk * 1024;
    for (int c = 0; c < 32; ++c) {
      const int d = lane;
      float val = ZZ[c * 32 + d] - mu[kk * 32 + c] * Sz[d]
                - mu[kk * 32 + d] * Sz[c]
                + (float)HW * mu[kk * 32 + c] * mu[kk * 32 + d];
      if (c == d) val += fmaxf(val, EPSF);
      cvk[c * 32 + d] = val;
      fr += val * val;
    }
    for (int o = 16; o > 0; o >>= 1) fr += __shfl_xor(fr, o, 32);
    const float inv = 1.0f / (EPSF + sqrtf(fr));
    for (int c = 0; c < 32; ++c) cvk[c * 32 + lane] *= inv;
  }
  __syncthreads();

  for (int j = 0; j < 32; ++j) {                     // Cholesky (right-looking)
    if (kk < NK) {
      const float s = sqrtf(cv[kk * 1024 + j * 33]);
      if (lane == j) cv[kk * 1024 + j * 33] = s;
      else if (lane > j) cv[kk * 1024 + lane * 32 + j] /= s;
    }
    __syncthreads();
    if (kk < NK && lane > j) {
      const float lij = cv[kk * 1024 + lane * 32 + j];
      for (int d = j + 1; d <= lane; ++d)
        cv[kk * 1024 + lane * 32 + d] -= lij * cv[kk * 1024 + d * 32 + j];
    }
    __syncthreads();
  }

  if (kk < NK) {                                     // A = L^-1 (col per lane)
    const float* Lk = cv + kk * 1024;
    float* Ak = Ai + kk * 1024;
    const int col = lane;
    for (int r = 0; r < 32; ++r) {
      if (r < col) { Ak[r * 32 + col] = 0.f; continue; }
      float acc = (r == col) ? 1.f : 0.f;
      for (int p = col; p < r; ++p) acc -= Lk[r * 32 + p] * Ak[p * 32 + col];
      Ak[r * 32 + col] = acc / Lk[r * 33];
    }
  }
  __syncthreads();

  if (kk < NK) {                                     // v = A mu, logdet, const
    const float* Ak = Ai + kk * 1024;
    float vv = 0.f;
    for (int c = 0; c <= lane; ++c) vv += Ak[lane * 32 + c] * mu[kk * 32 + c];
    float ld = logf(cv[kk * 1024 + lane * 33]);
    for (int o = 16; o > 0; o >>= 1) ld += __shfl_xor(ld, o, 32);
    const size_t base = (size_t)(b * NK + kk);
    for (int r = 0; r < 32; ++r)
      Aout[base * 1024 + r * 32 + lane] = Ak[r * 32 + lane];
    vout[base * 32 + lane] = vv;
    if (lane == 0) {
      const float phi = Sg[kk] / (float)HW;
      cst[base] = -0.5f * (32.0f * LOG2PI) - 0.5f * (2.0f * ld) + phi;
    }
  }
}

// ---------------------------------------------------------------------------
// K3: maha + streaming logsumexp. grid = 8 batches x 32, block = 256.
// Per 16-pixel tile: D = A_k(32x32) * Z(32x16) with triangular skip
// (4 + 8 K-chunks), maha = ||D - v||^2 folded over rows via shfl_xor(16).
// ---------------------------------------------------------------------------
__global__ __launch_bounds__(256) void k3_maha(const float* __restrict__ z,
                                               const float* __restrict__ Aglob,
                                               const float* __restrict__ vglob,
                                               float* __restrict__ lseout) {
  __shared__ float Al[NK * 1024];
  __shared__ float vl[NK * 32];
  __shared__ float lsered[NK * 8 * 16 * 2];
  const int b = blockIdx.x >> 5, sub = blockIdx.x & 31;
  const int wave = threadIdx.x >> 5, lane = threadIdx.x & 31;
  const int n = lane & 15, lh = lane >> 4;
  for (int e = threadIdx.x; e < NK * 1024; e += 256)
    Al[e] = Aglob[(size_t)b * NK * 1024 + e];
  for (int e = threadIdx.x; e < NK * 32; e += 256)
    vl[e] = vglob[(size_t)b * NK * 32 + e];
  __syncthreads();

  const float* zb = z + (size_t)b * NC * HW;
  float m[NK], s[NK];
#pragma unroll
  for (int k = 0; k < NK; ++k) { m[k] = -INFINITY; s[k] = 0.f; }

  const int pix0 = sub * 2048 + wave * 256;
  for (int t = 0; t < 16; ++t) {
    const int pb = pix0 + t * 16;
    v2f bf[8];                       // B frags: ch kc*4 + lh*2 {,+1}, px pb+n
#pragma unroll
    for (int kc = 0; kc < 8; ++kc) {
      const int ch = kc * 4 + lh * 2;
      bf[kc].x = zb[(size_t)ch * HW + pb + n];
      bf[kc].y = zb[(size_t)(ch + 1) * HW + pb + n];
    }
#pragma unroll
    for (int k = 0; k < NK; ++k) {
      const float* Ak = Al + k * 1024;
      v8f d0 = {}, d1 = {};
#pragma unroll
      for (int kc = 0; kc < 4; ++kc) {          // rows 0-15: cols 0-15 only
        const int col = kc * 4 + lh * 2;
        v2f af = {Ak[n * 32 + col], Ak[n * 32 + col + 1]};
        d0 = wmma4(af, bf[kc], d0);
      }
#pragma unroll
      for (int kc = 0; kc < 8; ++kc) {          // rows 16-31: all cols
        const int col = kc * 4 + lh * 2;
        v2f af = {Ak[(16 + n) * 32 + col], Ak[(16 + n) * 32 + col + 1]};
        d1 = wmma4(af, bf[kc], d1);
      }
      float prt = 0.f;
#pragma unroll
      for (int i = 0; i < 8; ++i) {
        const float y0 = d0[i] - vl[k * 32 + lh * 8 + i];
        const float y1 = d1[i] - vl[k * 32 + 16 + lh * 8 + i];
        prt += y0 * y0 + y1 * y1;
      }
      const float maha = prt + __shfl_xor(prt, 16, 32);
      const float x = -0.5f * maha;             // const added in K4
      if (x > m[k]) { s[k] = s[k] * expf(m[k] - x) + 1.f; m[k] = x; }
      else          { s[k] += expf(x - m[k]); }
    }
  }

  if (lh == 0) {                                // lanes 0-15 hold valid pairs
#pragma unroll
    for (int k = 0; k < NK; ++k) {
      const int idx = ((k * 8 + wave) * 16 + n) * 2;
      lsered[idx] = m[k];
      lsered[idx + 1] = s[k];
    }
  }
  __syncthreads();
  if (threadIdx.x < NK) {                       // fixed order -> deterministic
    const int k = threadIdx.x;
    float M = -INFINITY, S = 0.f;
    for (int e = 0; e < 128; ++e) {
      const float mi = lsered[(k * 128 + e) * 2];
      const float si = lsered[(k * 128 + e) * 2 + 1];
      if (mi > M) { S = S * expf(M - mi) + si; M = mi; }
      else        { S += si * expf(mi - M); }
    }
    const int o = (b * NK + k) * 32 + sub;
    lseout[o * 2] = M;
    lseout[o * 2 + 1] = S;
  }
}

// ---------------------------------------------------------------------------
// K4: final scalar energy.
// ---------------------------------------------------------------------------
__global__ __launch_bounds__(64) void k4_final(const float* __restrict__ lsein,
                                               const float* __restrict__ cst,
                                               float* __restrict__ out) {
  __shared__ float acc[64];
  const int tid = threadIdx.x;
  float ls = 0.f;
  if (tid < NB * NK) {
    float M = -INFINITY, S = 0.f;
    for (int sub = 0; sub < 32; ++sub) {
      const float mi = lsein[(tid * 32 + sub) * 2];
      const float si = lsein[(tid * 32 + sub) * 2 + 1];
      if (mi > M) { S = S * expf(M - mi) + si; M = mi; }
      else        { S += si * expf(mi - M); }
    }
    ls = cst[tid] + M + logf(S);
  }
  acc[tid] = ls;
  __syncthreads();
  if (tid == 0) {
    float t = 0.f;
    for (int i = 0; i < NB * NK; ++i) t += acc[i];
    out[0] = -(t / (float)(NB * NK)) / (float)HW;
  }
}

// ---------------------------------------------------------------------------
extern "C" void kernel_launch(void* const* d_in, const int* in_sizes, int n_in,
                              void* d_out, int out_size, void* d_ws,
                              size_t ws_size, hipStream_t stream) {
  (void)in_sizes; (void)n_in; (void)out_size; (void)ws_size;
  const float* z = (const float*)d_in[0];   // [8,32,256,256] f32
  const float* g = (const float*)d_in[1];   // [8,5,256,256]  f32
  float* ws = (float*)d_ws;
  // workspace layout (floats): needs ~573 KB total
  float* part = ws;                  // 64 wg * 1536
  float* Aout = part + 98304;        // 40 * 1024
  float* vout = Aout + 40960;        // 40 * 32
  float* cst  = vout + 1280;         // 40
  float* lse  = cst + 40;            // 40 * 32 * 2

  k1_stats<<<dim3(64), dim3(256), 0, stream>>>(z, g, part);
  k2_prep <<<dim3(8),  dim3(256), 0, stream>>>(part, Aout, vout, cst);
  k3_maha <<<dim3(256), dim3(256), 0, stream>>>(z, Aout, vout, lse);
  k4_final<<<dim3(1),  dim3(64),  0, stream>>>(lse, cst, (float*)d_out);
}